// LsqWeight_2b_4595615007450
// MI455X (gfx1250) — compile-verified
//
#include <hip/hip_runtime.h>
#include <hip/hip_bf16.h>
#include <stdint.h>

// LSQ+-style 2-bit dual-scale quantizer, forward pass.
// C=4096 channels, N=4096 elems/channel, 9 levels/channel.
// Memory-bound streaming kernel using CDNA5 async global->LDS double buffering.

#define CH_N 4096 // elements per channel row
#define STAGE_ELEMS 1024 // 256 threads * 4 floats per stage
#define N_STAGES 4 // 4096 / 1024

__device__ __forceinline__ float quant9(float xv, const float (&lv)[9]) {
    // distances to all 9 levels + argmin (first-min wins, matching jnp.argmin)
    float d[9];
    d[0] = fabsf(xv - lv[0]);
    float dmin = d[0];
    float hard = lv[0];
#pragma unroll
    for (int k = 1; k < 9; ++k) {
        d[k] = fabsf(xv - lv[k]);
        if (d[k] < dmin) { dmin = d[k]; hard = lv[k]; }
    }
    // softmax(-alpha*d) over k, max-subtracted (jax.nn.softmax subtracts max).
    // exp(-alpha*(d-dmin)) == exp2(K*(d-dmin)), K = -alpha*log2(e), alpha=10.
    const float K = -14.426950408889634f;
    float sw = 0.0f, swl = 0.0f;
#pragma unroll
    for (int k = 0; k < 9; ++k) {
        float w = __builtin_amdgcn_exp2f(K * (d[k] - dmin)); // v_exp_f32 (TRANS pipe)
        sw += w;
        swl = fmaf(w, lv[k], swl);
    }
    float soft = swl * __builtin_amdgcn_rcpf(sw); // v_rcp_f32
    // STE forward: stop_gradient(hard - soft) + soft
    return (hard - soft) + soft;
}

__global__ __launch_bounds__(256, 2) void lsq2b_fwd(const float* __restrict__ x,
                                                    const float* __restrict__ sH,
                                                    const float* __restrict__ sL,
                                                    float* __restrict__ out) {
    __shared__ float lds[2 * STAGE_ELEMS]; // 8 KB double buffer

    const int c = blockIdx.x;
    const int tid = threadIdx.x;

    const float h = sH[c];
    const float l = sL[c];
    // 9 levels = sH*c0 + sL*c1 over COEFFS order [-1,-1],[-1,0],...,[1,1]
    float lv[9];
    lv[0] = -h - l; lv[1] = -h;   lv[2] = -h + l;
    lv[3] = -l;     lv[4] = 0.0f; lv[5] = l;
    lv[6] = h - l;  lv[7] = h;    lv[8] = h + l;

    // uniform 64-bit global base of this channel's row (SGPR pair for GVS mode)
    const unsigned long long gbase =
        (unsigned long long)(uintptr_t)x + (unsigned long long)c * (CH_N * 4ull);
    // LDS byte address of the shared buffer (flat addr low 32 bits == LDS offset)
    const unsigned int lbase = (unsigned int)(size_t)(&lds[0]);

    // Issue stage 0 async load: each lane pulls its own 16 bytes into LDS.
    {
        unsigned int voff  = (unsigned int)tid * 16u;            // byte offset in row
        unsigned int laddr = lbase + (unsigned int)tid * 16u;    // LDS dest byte addr
        asm volatile("global_load_async_to_lds_b128 %0, %1, %2"
                     :: "v"(laddr), "v"(voff), "s"(gbase)
                     : "memory");
    }

    float* orow = out + (size_t)c * CH_N;

#pragma unroll
    for (int s = 0; s < N_STAGES; ++s) {
        if (s < N_STAGES - 1) {
            // prefetch next stage, then wait until only that one is outstanding
            unsigned int voff  = (unsigned int)(s + 1) * (STAGE_ELEMS * 4u)
                               + (unsigned int)tid * 16u;
            unsigned int laddr = lbase
                               + (unsigned int)((s + 1) & 1) * (STAGE_ELEMS * 4u)
                               + (unsigned int)tid * 16u;
            asm volatile("global_load_async_to_lds_b128 %0, %1, %2"
                         :: "v"(laddr), "v"(voff), "s"(gbase)
                         : "memory");
            asm volatile("s_wait_asynccnt 1" ::: "memory");
        } else {
            asm volatile("s_wait_asynccnt 0" ::: "memory");
        }

        // ds_load_b128: lane reads back exactly the 16B it async-loaded
        const float4 v =
            *reinterpret_cast<const float4*>(&lds[(s & 1) * STAGE_ELEMS + tid * 4]);

        float4 r;
        r.x = quant9(v.x, lv);
        r.y = quant9(v.y, lv);
        r.z = quant9(v.z, lv);
        r.w = quant9(v.w, lv);

        // global_store_b128
        *reinterpret_cast<float4*>(orow + s * STAGE_ELEMS + tid * 4) = r;
    }
}

extern "C" void kernel_launch(void* const* d_in, const int* in_sizes, int n_in,
                              void* d_out, int out_size, void* d_ws, size_t ws_size,
                              hipStream_t stream) {
    (void)n_in; (void)d_ws; (void)ws_size; (void)out_size;

    const float* x  = (const float*)d_in[0];
    const float* sH = (const float*)d_in[1];
    const float* sL = (const float*)d_in[2];
    float* out = (float*)d_out;

    const int C = in_sizes[1];              // 4096 channels (sH has C elements)
    // One block per channel; 256 threads; 4-stage async double-buffered pipeline.
    lsq2b_fwd<<<C, 256, 0, stream>>>(x, sH, sL, out);

    // Tuple tail: sH_s and sL_s forward values are identity copies.
    const size_t wq_elems = (size_t)in_sizes[0];
    hipMemcpyAsync(out + wq_elems, sH, (size_t)in_sizes[1] * sizeof(float),
                   hipMemcpyDeviceToDevice, stream);
    hipMemcpyAsync(out + wq_elems + (size_t)in_sizes[1], sL,
                   (size_t)in_sizes[2] * sizeof(float),
                   hipMemcpyDeviceToDevice, stream);
}